// GINConv_attn_20641612824581
// MI455X (gfx1250) — compile-verified
//
#include <hip/hip_runtime.h>
#include <hip/hip_bf16.h>

typedef __attribute__((ext_vector_type(2))) float v2f;
typedef __attribute__((ext_vector_type(8))) float v8f;

#define NEG_SLOPE 0.2f
#define D 128

// ---------------------------------------------------------------------------
// Kernel 1: proj[N,128] = feat[N,128] @ W[128,128]^T via V_WMMA_F32_16X16X4_F32
// 8 waves / block, each wave owns one 16x16 output tile (wave w -> cols w*16..).
// grid.x covers M in 16-row tiles. K loop: 128/4 = 32 WMMA issues per wave.
// ---------------------------------------------------------------------------
__global__ __launch_bounds__(256) void proj_gemm_wmma(
    const float* __restrict__ feat, const float* __restrict__ W,
    float* __restrict__ proj, int N) {
  const int lane = threadIdx.x & 31;
  const int wave = threadIdx.x >> 5;
  const int half = lane >> 4;   // 0: lanes 0-15, 1: lanes 16-31
  const int l15  = lane & 15;
  const int m0 = blockIdx.x * 16;
  const int n0 = wave * 16;

  int mrow = m0 + l15;
  if (mrow >= N) mrow = N - 1;              // clamp so EXEC stays all-ones
  const int ncol = n0 + l15;

  const float* __restrict__ arow = feat + (size_t)mrow * D;
  const float* __restrict__ brow = W    + (size_t)ncol * D;

  v8f acc = {};
#pragma unroll 4
  for (int k0 = 0; k0 < D; k0 += 4) {
    const int kb = k0 + (half << 1);
    v2f a; a.x = arow[kb]; a.y = arow[kb + 1];   // A: M=l15, K=kb..kb+1
    v2f b; b.x = brow[kb]; b.y = brow[kb + 1];   // B: N=l15, K=kb..kb+1 (W^T)
    acc = __builtin_amdgcn_wmma_f32_16x16x4_f32(
        /*neg_a=*/false, a, /*neg_b=*/false, b,
        /*c_mod=*/(short)0, acc, /*reuse_a=*/false, /*reuse_b=*/false);
  }

  // C/D layout: VGPR r -> M = r + 8*half, N = l15
  const int mbase = m0 + half * 8;
  float* __restrict__ cptr = proj + (size_t)mbase * D + n0 + l15;
  if (m0 + 16 <= N) {
    // wave-uniform fast path: straight-line stores, no per-element exec games
#pragma unroll
    for (int r = 0; r < 8; ++r) cptr[(size_t)r * D] = acc[r];
  } else {
#pragma unroll
    for (int r = 0; r < 8; ++r)
      if (mbase + r < N) cptr[(size_t)r * D] = acc[r];
  }
}

// ---------------------------------------------------------------------------
// Kernel 2: fused edge phase. One wave32 per edge; lane owns a float4 chunk.
//   e    = sum(leaky_relu(el+er)*a);  gate = sigmoid(el.er);  w = sigmoid(e*gate)
//   neigh[dst] += feat[src]*w (L2-resident f32 atomics);  deg[dst] += 1
// ---------------------------------------------------------------------------
__global__ __launch_bounds__(256) void edge_kernel(
    const float* __restrict__ proj, const float* __restrict__ feat,
    const float* __restrict__ attn, const int* __restrict__ src,
    const int* __restrict__ dst, float* __restrict__ neigh,
    float* __restrict__ deg, int E) {
  const int edge = blockIdx.x * 8 + (threadIdx.x >> 5);
  const int lane = threadIdx.x & 31;
  if (edge >= E) return;

  const int s = src[edge];
  const int d = dst[edge];

  const float4 el = *(const float4*)(proj + (size_t)s * D + lane * 4);
  const float4 er = *(const float4*)(proj + (size_t)d * D + lane * 4);
  const float4 av = *(const float4*)(attn + lane * 4);

  float ep = 0.f, gp = 0.f;
  float t;
  t = el.x + er.x; ep += (t > 0.f ? t : NEG_SLOPE * t) * av.x; gp += el.x * er.x;
  t = el.y + er.y; ep += (t > 0.f ? t : NEG_SLOPE * t) * av.y; gp += el.y * er.y;
  t = el.z + er.z; ep += (t > 0.f ? t : NEG_SLOPE * t) * av.z; gp += el.z * er.z;
  t = el.w + er.w; ep += (t > 0.f ? t : NEG_SLOPE * t) * av.w; gp += el.w * er.w;

#pragma unroll
  for (int off = 16; off; off >>= 1) {       // wave32 butterfly reduce
    ep += __shfl_xor(ep, off, 32);
    gp += __shfl_xor(gp, off, 32);
  }

  const float gate = 1.f / (1.f + __expf(-gp));
  const float wgt  = 1.f / (1.f + __expf(-(ep * gate)));

  const float4 f = *(const float4*)(feat + (size_t)s * D + lane * 4);
  float* base = neigh + (size_t)d * D + lane * 4;
  atomicAdd(base + 0, f.x * wgt);
  atomicAdd(base + 1, f.y * wgt);
  atomicAdd(base + 2, f.z * wgt);
  atomicAdd(base + 3, f.w * wgt);
  if (lane == 0) atomicAdd(deg + d, 1.0f);
}

// ---------------------------------------------------------------------------
// Kernel 3: rst = (1+eps)*feat + neigh / max(deg, 1)   (float4 streaming)
// ---------------------------------------------------------------------------
__global__ __launch_bounds__(256) void finalize_kernel(
    const float* __restrict__ feat, const float* __restrict__ neigh,
    const float* __restrict__ deg, const float* __restrict__ eps,
    float* __restrict__ out, int N) {
  const int i = blockIdx.x * blockDim.x + threadIdx.x;   // float4 index
  const int total4 = N * (D / 4);
  if (i >= total4) return;
  const int row = i >> 5;                                // / (D/4)
  float dg = deg[row];
  dg = dg > 1.f ? dg : 1.f;
  const float inv = 1.f / dg;
  const float scale = 1.f + eps[0];
  const float4 f = ((const float4*)feat)[i];
  const float4 nb = ((const float4*)neigh)[i];
  float4 o;
  o.x = scale * f.x + nb.x * inv;
  o.y = scale * f.y + nb.y * inv;
  o.z = scale * f.z + nb.z * inv;
  o.w = scale * f.w + nb.w * inv;
  ((float4*)out)[i] = o;
}

extern "C" void kernel_launch(void* const* d_in, const int* in_sizes, int n_in,
                              void* d_out, int out_size, void* d_ws, size_t ws_size,
                              hipStream_t stream) {
  const float* feat = (const float*)d_in[0];
  const float* W    = (const float*)d_in[1];
  const float* attn = (const float*)d_in[2];
  const float* eps  = (const float*)d_in[3];
  const int*   src  = (const int*)d_in[4];
  const int*   dst  = (const int*)d_in[5];

  const int N = in_sizes[0] / D;
  const int E = in_sizes[4];

  float* proj  = (float*)d_ws;                  // N*D floats
  float* neigh = proj + (size_t)N * D;          // N*D floats
  float* deg   = neigh + (size_t)N * D;         // N floats

  // zero the accumulators (graph-capture-safe async memset)
  hipMemsetAsync(neigh, 0, ((size_t)N * D + N) * sizeof(float), stream);

  proj_gemm_wmma<<<(N + 15) / 16, 256, 0, stream>>>(feat, W, proj, N);
  edge_kernel<<<(E + 7) / 8, 256, 0, stream>>>(proj, feat, attn, src, dst,
                                               neigh, deg, E);
  const int total4 = N * (D / 4);
  finalize_kernel<<<(total4 + 255) / 256, 256, 0, stream>>>(feat, neigh, deg,
                                                            eps, (float*)d_out, N);
}